// GCNModel_60129542735
// MI455X (gfx1250) — compile-verified
//
#include <hip/hip_runtime.h>

typedef __attribute__((ext_vector_type(2))) float v2f;
typedef __attribute__((ext_vector_type(8))) float v8f;

// ---------------------------------------------------------------------------
// CDNA5 async global->LDS copy (16B per lane), tracked on ASYNCcnt.
// VDST operand = wave-relative LDS byte offset; generic LDS pointers on
// AMDGPU are {shared_aperture_hi, lds_offset32}, so low 32 bits suffice.
// ---------------------------------------------------------------------------
__device__ __forceinline__ void async_copy_b128(void* lds_dst, const void* gsrc) {
    unsigned lds_off = (unsigned)(size_t)lds_dst;
    asm volatile("global_load_async_to_lds_b128 %0, %1, off"
                 :: "v"(lds_off), "v"(gsrc)
                 : "memory");
}

__device__ __forceinline__ void wait_asynccnt0() {
    asm volatile("s_wait_asynccnt 0x0" ::: "memory");
}

// ---------------------------------------------------------------------------
// Degree / normalization kernels
// ---------------------------------------------------------------------------
__global__ void k_init_deg(float* deg, int n) {
    int i = blockIdx.x * blockDim.x + threadIdx.x;
    if (i < n) deg[i] = 1.0f;  // self-loop
}

__global__ void k_deg_scatter(const long long* __restrict__ dst, float* deg, long long E) {
    long long e = (long long)blockIdx.x * blockDim.x + threadIdx.x;
    if (e < E) atomicAdd(&deg[dst[e]], 1.0f);
}

__global__ void k_rsqrt_inplace(float* deg, int n) {
    int i = blockIdx.x * blockDim.x + threadIdx.x;
    if (i < n) deg[i] = __frsqrt_rn(deg[i]);  // deg >= 1 always
}

// ---------------------------------------------------------------------------
// f32 WMMA GEMM:  out[N x NOUT] = A[N x 128] @ W[128 x NOUT]
// Block: 256 threads = 8 waves. Each wave owns one 16x16 output tile.
//   NOUT=128 -> 8 column tiles, 16 rows/block
//   NOUT=64  -> 4 column tiles, 32 rows/block
// Staging: global_load_async_to_lds_b128 (ASYNCcnt), then
// V_WMMA_F32_16X16X4_F32, K swept in steps of 4 (32 MMAs per tile).
// ---------------------------------------------------------------------------
template <int NOUT>
__global__ __launch_bounds__(256) void k_gemm_wmma(const float* __restrict__ A,
                                                   const float* __restrict__ W,
                                                   float* __restrict__ out, int N) {
    constexpr int KDIM = 128;
    constexpr int CT = NOUT / 16;      // column tiles per block
    constexpr int RT = 8 / CT;         // row sub-tiles per block
    constexpr int ROWS = 16 * RT;      // rows per block

    __shared__ float sW[KDIM * NOUT];  // 64KB (NOUT=128) / 32KB (NOUT=64)
    __shared__ float sA[ROWS * KDIM];  // 8KB / 16KB

    const int tid = threadIdx.x;
    const int rowBase = blockIdx.x * ROWS;

    // Stage W via async copies (16B per lane per issue)
    for (int i = tid * 4; i < KDIM * NOUT; i += 256 * 4) {
        async_copy_b128(&sW[i], &W[i]);
    }
    // Stage A tile (clamp row for tail safety; N is a multiple of ROWS here)
    for (int i = tid * 4; i < ROWS * KDIM; i += 256 * 4) {
        int r = i / KDIM, c = i % KDIM;
        int gr = rowBase + r;
        if (gr >= N) gr = N - 1;
        async_copy_b128(&sA[i], &A[(size_t)gr * KDIM + c]);
    }
    wait_asynccnt0();
    __syncthreads();

    const int wave = tid >> 5;
    const int lane = tid & 31;
    const int col0 = (wave % CT) * 16;
    const int rsub = (wave / CT) * 16;
    const int half = lane >> 4;   // 0: K pair {0,1}; 1: K pair {2,3}
    const int m    = lane & 15;

    v8f acc = {};
#pragma unroll
    for (int k = 0; k < KDIM; k += 4) {
        // A fragment (16x4): lane half selects K sub-pair
        const float* ap = &sA[(rsub + m) * KDIM + k + 2 * half];
        v2f a;
        a.x = ap[0];
        a.y = ap[1];
        // B fragment (4x16): VGPR0 holds K = k+2*half, VGPR1 holds K = k+2*half+1
        v2f b;
        b.x = sW[(k + 2 * half + 0) * NOUT + col0 + m];
        b.y = sW[(k + 2 * half + 1) * NOUT + col0 + m];
        acc = __builtin_amdgcn_wmma_f32_16x16x4_f32(false, a, false, b,
                                                    (short)0, acc, false, false);
    }

    // C/D layout: VGPR j -> row j (lanes 0-15) / row j+8 (lanes 16-31), col = m
#pragma unroll
    for (int j = 0; j < 8; ++j) {
        int r = rowBase + rsub + j + 8 * half;
        if (r < N) out[(size_t)r * NOUT + col0 + m] = acc[j];
    }
}

// ---------------------------------------------------------------------------
// Aggregation: out = D^-1/2 (A+I) D^-1/2 feat + bias
// ---------------------------------------------------------------------------
template <int C>
__global__ void k_agg_init(const float* __restrict__ feat, const float* __restrict__ dinv,
                           const float* __restrict__ bias, float* __restrict__ out, int N) {
    size_t i = (size_t)blockIdx.x * blockDim.x + threadIdx.x;
    size_t total = (size_t)N * C;
    if (i >= total) return;
    int node = (int)(i / C);
    int c = (int)(i % C);
    float dv = dinv[node];
    out[i] = feat[i] * dv * dv + bias[c];  // self-loop term + bias folded in
}

template <int C>
__global__ void k_agg_scatter(const float* __restrict__ feat, const float* __restrict__ dinv,
                              const long long* __restrict__ src, const long long* __restrict__ dst,
                              float* out, long long E) {
    const int wavesPerBlock = blockDim.x >> 5;
    long long e = (long long)blockIdx.x * wavesPerBlock + (threadIdx.x >> 5);
    if (e >= E) return;
    int lane = threadIdx.x & 31;
    long long s = src[e];
    long long d = dst[e];
    const float* frow = feat + (size_t)s * C;
    // Kick the row fetch early (global_prefetch_b8) while dinv loads resolve.
    __builtin_prefetch(frow + lane * 2, 0, 1);
    float nrm = dinv[s] * dinv[d];
    float* orow = out + (size_t)d * C;
#pragma unroll
    for (int c = lane; c < C; c += 32) {
        atomicAdd(&orow[c], frow[c] * nrm);
    }
}

__global__ void k_relu_inplace(float* x, size_t total) {
    size_t i = (size_t)blockIdx.x * blockDim.x + threadIdx.x;
    if (i < total) x[i] = fmaxf(x[i], 0.0f);
}

// ---------------------------------------------------------------------------
// Launch
// ---------------------------------------------------------------------------
extern "C" void kernel_launch(void* const* d_in, const int* in_sizes, int n_in,
                              void* d_out, int out_size, void* d_ws, size_t ws_size,
                              hipStream_t stream) {
    const float* x       = (const float*)d_in[0];
    const long long* ei  = (const long long*)d_in[1];
    const float* W1      = (const float*)d_in[2];
    const float* b1      = (const float*)d_in[3];
    const float* W2      = (const float*)d_in[4];
    const float* b2      = (const float*)d_in[5];
    float* out           = (float*)d_out;

    const int  N = in_sizes[0] / 128;          // 100000
    const long long E = in_sizes[1] / 2;       // 1600000
    const long long* src = ei;
    const long long* dst = ei + E;

    float* ws = (float*)d_ws;
    size_t nAlign = ((size_t)N + 255) & ~(size_t)255;
    float* dinv = ws;                          // N floats
    float* buf1 = ws + nAlign;                 // N*128 (xW1, later hW2)
    float* buf2 = buf1 + (size_t)N * 128;      // N*128 (agg1 / h)

    const int T = 256;

    // 1) degrees -> dinv
    k_init_deg<<<(N + T - 1) / T, T, 0, stream>>>(dinv, N);
    k_deg_scatter<<<(unsigned)((E + T - 1) / T), T, 0, stream>>>(dst, dinv, E);
    k_rsqrt_inplace<<<(N + T - 1) / T, T, 0, stream>>>(dinv, N);

    // 2) layer 1: xW1 (WMMA f32), aggregate, relu
    k_gemm_wmma<128><<<(N + 15) / 16, T, 0, stream>>>(x, W1, buf1, N);
    {
        size_t total = (size_t)N * 128;
        k_agg_init<128><<<(unsigned)((total + T - 1) / T), T, 0, stream>>>(buf1, dinv, b1, buf2, N);
        unsigned eb = (unsigned)((E + 7) / 8);  // 8 waves/block, 1 wave/edge
        k_agg_scatter<128><<<eb, T, 0, stream>>>(buf1, dinv, src, dst, buf2, E);
        k_relu_inplace<<<(unsigned)((total + T - 1) / T), T, 0, stream>>>(buf2, total);
    }

    // 3) layer 2: hW2 (WMMA f32), aggregate into d_out
    k_gemm_wmma<64><<<(N + 31) / 32, T, 0, stream>>>(buf2, W2, buf1, N);
    {
        size_t total = (size_t)N * 64;
        k_agg_init<64><<<(unsigned)((total + T - 1) / T), T, 0, stream>>>(buf1, dinv, b2, out, N);
        unsigned eb = (unsigned)((E + 7) / 8);
        k_agg_scatter<64><<<eb, T, 0, stream>>>(buf1, dinv, src, dst, out, E);
    }
}